// FocalLoss_1743756722408
// MI455X (gfx1250) — compile-verified
//
#include <hip/hip_runtime.h>
#include <hip/hip_bf16.h>
#include <math.h>

// Focal loss, MI455X (gfx1250, wave32).
// Pass 1: wave-per-row streaming softmax stats (HBM-bound: 1.05GB / 23.3TB/s ~= 45us floor).
//         Cross-lane reductions use DS_SWIZZLE_B32 xor-mode (1 DS op/step vs ~5 ops for shfl).
// Pass 2: batched float4 accumulation (4 loads in flight, no per-load waits), then a
//         V_WMMA_F32_16X16X4_F32 ones-A fold (D[m][n] = sum_k B[k][n]) + 4 swizzle steps.

typedef float v4f __attribute__((ext_vector_type(4)));
typedef float v2f __attribute__((ext_vector_type(2)));
typedef float v8f __attribute__((ext_vector_type(8)));

#define C_DIM          1000
#define F4_DIM         250     // C_DIM / 4 float4's per row
#define ITERS          8       // ceil(250 / 32) float4 loads per lane
#define ROWS_PER_BLOCK 16      // one wave32 per row
#define THREADS_1      (ROWS_PER_BLOCK * 32)

// ds_swizzle_b32 group-of-32 xor-shuffle: offset = {1'b0, xor[14:10], or[9:5], and[4:0]}
#define SWZ_XOR(dst, src, xmask)                                               \
    dst = __int_as_float(__builtin_amdgcn_ds_swizzle(                          \
        __float_as_int(src), ((xmask) << 10) | 0x1f))

__device__ __forceinline__ float wave_max32(float m) {
    float t;
    SWZ_XOR(t, m, 1);  m = fmaxf(m, t);
    SWZ_XOR(t, m, 2);  m = fmaxf(m, t);
    SWZ_XOR(t, m, 4);  m = fmaxf(m, t);
    SWZ_XOR(t, m, 8);  m = fmaxf(m, t);
    SWZ_XOR(t, m, 16); m = fmaxf(m, t);
    return m;
}

__device__ __forceinline__ float wave_sum32(float s) {
    float t;
    SWZ_XOR(t, s, 1);  s += t;
    SWZ_XOR(t, s, 2);  s += t;
    SWZ_XOR(t, s, 4);  s += t;
    SWZ_XOR(t, s, 8);  s += t;
    SWZ_XOR(t, s, 16); s += t;
    return s;
}

__global__ __launch_bounds__(THREADS_1) void focal_rows_kernel(
    const float* __restrict__ x, const int* __restrict__ tgt,
    float* __restrict__ partial, int nrows)
{
    __shared__ float wave_loss[ROWS_PER_BLOCK];
    const int lane = threadIdx.x & 31;
    const int wave = threadIdx.x >> 5;
    const int row  = blockIdx.x * ROWS_PER_BLOCK + wave;

    float loss = 0.0f;
    if (row < nrows) {
        const float* rp = x + (size_t)row * C_DIM;   // row base: 4000B -> 16B aligned

        // Single memory pass: whole row resident in registers (32 floats/lane).
        v4f v[ITERS];
        float m = -INFINITY;
#pragma unroll
        for (int j = 0; j < ITERS; ++j) {
            const int f = j * 32 + lane;             // float4 index, coalesced b128 loads
            if (f < F4_DIM) {
                v[j] = __builtin_nontemporal_load((const v4f*)rp + f);
            } else {
                v[j] = (v4f){-INFINITY, -INFINITY, -INFINITY, -INFINITY};
            }
            m = fmaxf(m, fmaxf(fmaxf(v[j][0], v[j][1]), fmaxf(v[j][2], v[j][3])));
        }
        m = wave_max32(m);

        // exp-sum from registers; padded -INF entries contribute exp(-inf)=0
        float s = 0.0f;
#pragma unroll
        for (int j = 0; j < ITERS; ++j) {
            s += __expf(v[j][0] - m) + __expf(v[j][1] - m)
               + __expf(v[j][2] - m) + __expf(v[j][3] - m);
        }
        s = wave_sum32(s);

        if (lane == 0) {
            const int   t     = tgt[row];
            const float xt    = rp[t];                    // L2-hot reload (192MB L2)
            const float logpt = xt - m - __logf(s);       // log_softmax at target
            const float pt    = __expf(logpt);
            const float w     = (t > 0) ? 0.25f : 0.75f;  // ALPHA / 1-ALPHA
            const float om    = 1.0f - pt;
            loss = -(om * om) * (w * logpt);              // GAMMA = 2
        }
    }
    if (lane == 0) wave_loss[wave] = loss;
    __syncthreads();
    if (threadIdx.x == 0) {
        float b = 0.0f;
#pragma unroll
        for (int i = 0; i < ROWS_PER_BLOCK; ++i) b += wave_loss[i];
        partial[blockIdx.x] = b;
    }
}

__global__ __launch_bounds__(256) void reduce_mean_kernel(
    const float* __restrict__ partial, float* __restrict__ out,
    int n, float inv_n)
{
    __shared__ float wsum[8];
    const int lane = threadIdx.x & 31;
    const int wave = threadIdx.x >> 5;

    // Batched accumulation: 4 independent float4 streams so the compiler can
    // clause 4 b128 loads per wait instead of one load+wait round-trip each.
    const v4f* p4 = (const v4f*)partial;
    const int  n4 = n >> 2;
    v4f a0 = {0.f, 0.f, 0.f, 0.f}, a1 = a0, a2 = a0, a3 = a0;
    int i = threadIdx.x;
    for (; i + 768 < n4; i += 1024) {
        a0 += p4[i];
        a1 += p4[i + 256];
        a2 += p4[i + 512];
        a3 += p4[i + 768];
    }
    for (; i < n4; i += 256) a0 += p4[i];
    const v4f at = (a0 + a1) + (a2 + a3);
    float s = (at[0] + at[1]) + (at[2] + at[3]);
    for (int j = (n4 << 2) + (int)threadIdx.x; j < n; j += 256) s += partial[j]; // n%4 tail

    // WMMA cross-lane fold (uniform control flow here -> EXEC all-ones as required):
    // A = ones(16x4), B holds each lane's partial in one VGPR (other zeroed).
    // D[m][n] = sum_k B[k][n]: every column is the sum of a disjoint pair of lanes,
    // so summing row 0's 16 columns yields the full 32-lane sum (layout-robust
    // because A is all-ones and B's slot->(k,n) map is a bijection).
    v2f a; a[0] = 1.0f; a[1] = 1.0f;
    v2f b; b[0] = s;    b[1] = 0.0f;
    v8f c = {};
    v8f d = __builtin_amdgcn_wmma_f32_16x16x4_f32(
        /*neg_a=*/false, a, /*neg_b=*/false, b,
        /*c_mod=*/(short)0, c, /*reuse_a=*/false, /*reuse_b=*/false);

    float r = d[0], t;                 // lane L: column sum for N = L%16
    SWZ_XOR(t, r, 1); r += t;
    SWZ_XOR(t, r, 2); r += t;
    SWZ_XOR(t, r, 4); r += t;
    SWZ_XOR(t, r, 8); r += t;

    if (lane == 0) wsum[wave] = r;
    __syncthreads();
    if (threadIdx.x == 0) {
        float tot = 0.0f;
#pragma unroll
        for (int k = 0; k < 8; ++k) tot += wsum[k];
        out[0] = tot * inv_n;          // SIZE_AVERAGE: mean over N rows
    }
}

extern "C" void kernel_launch(void* const* d_in, const int* in_sizes, int n_in,
                              void* d_out, int out_size, void* d_ws, size_t ws_size,
                              hipStream_t stream) {
    const float* x   = (const float*)d_in[0];   // [N, 1000] fp32 logits
    const int*   tgt = (const int*)d_in[1];     // [N] class ids
    float* out     = (float*)d_out;             // scalar mean loss
    float* partial = (float*)d_ws;              // one float per block

    const int N      = in_sizes[1];
    const int blocks = (N + ROWS_PER_BLOCK - 1) / ROWS_PER_BLOCK;  // 16384 for N=262144

    focal_rows_kernel<<<blocks, THREADS_1, 0, stream>>>(x, tgt, partial, N);
    reduce_mean_kernel<<<1, 256, 0, stream>>>(partial, out, blocks, 1.0f / (float)N);
}